// tensor1x1conv_TR_40596030882210
// MI455X (gfx1250) — compile-verified
//
#include <hip/hip_runtime.h>
#include <math.h>

typedef float v2f __attribute__((ext_vector_type(2)));
typedef float v8f __attribute__((ext_vector_type(8)));

#define HW    4096            // 64*64
#define CHW   (256 * 4096)    // C_IN * H * W
#define PIXS  292             // padded per-pixel LDS stride (floats); 292%64=36 -> conflict-free
#define MROW  18              // padded row stride inside a 16x16 block
#define YROW  20              // padded row stride of the output transpose tile
#define WAVES 2

// ---------------------------------------------------------------------------
// Pack both B operands pair-interleaved so each WMMA B-fragment is one b64 load.
//   Wp2 (float2)[t*2048 + m*16 + n] = { w_core[t, 2m,   n], w_core[t, 2m+1, n] }
//   Vp2 (float2)[m*256 + d]         = { V[2m, d], V[2m+1, d] },  V[e*16+g, d] = w_out[g, d, e]
// Output floats: Wp at ws[0 .. 65535], Vp at ws[65536 .. 131071].
// ---------------------------------------------------------------------------
__global__ __launch_bounds__(256) void repack_weights(const float* __restrict__ wcore,
                                                      const float* __restrict__ wout,
                                                      float* __restrict__ ws) {
    int idx = blockIdx.x * 256 + threadIdx.x;          // 0 .. 131071
    if (idx < 65536) {
        // w_core packing
        int lo   = idx & 1;
        int pair = idx >> 1;
        int n    = pair & 15;
        int m    = (pair >> 4) & 127;
        int t    = pair >> 11;
        int c    = 2 * m + lo;
        ws[idx] = wcore[t * 4096 + c * 16 + n];
    } else {
        // w_out packing
        int f    = idx - 65536;
        int lo   = f & 1;
        int rest = f >> 1;
        int d    = rest & 255;
        int m    = rest >> 8;
        int k    = 2 * m + lo;                         // k = e*16 + g
        int e    = k >> 4;
        int g    = k & 15;
        ws[idx] = wout[g * 4096 + d * 16 + e];
    }
}

// ---------------------------------------------------------------------------
// Main pass: per 16-pixel tile (one wave each):
//   GEMM1:  M[16px][256] = X[16px][256ch] * Wcore[256ch][256rk]   (fp32 WMMA)
//   square: P_px = M_px (16x16) @ M_px                            (fp32 WMMA)
//   GEMM2:  Y[16px][256] = P[16px][256] * Vpack[256][256]         (fp32 WMMA)
// Y written to d_out in [a][d][h*w] layout via an LDS transpose tile.
// ---------------------------------------------------------------------------
__global__ __launch_bounds__(WAVES * 32) void tr_main(const float* __restrict__ x,
                                                      const float* __restrict__ wp2,
                                                      const float* __restrict__ vp2,
                                                      float* __restrict__ out) {
    extern __shared__ float smem[];
    const int wave = threadIdx.x >> 5;
    const int lane = threadIdx.x & 31;
    const int half = lane >> 4;      // 0: lanes 0-15, 1: lanes 16-31
    const int ln   = lane & 15;

    float* Mw = smem + wave * (16 * PIXS);                   // 16 px, padded 16x16 blocks
    float* Yw = smem + WAVES * (16 * PIXS) + wave * (16 * YROW);

    const int tile = blockIdx.x * WAVES + wave;              // 0 .. 4095
    const int img  = tile >> 8;                              // batch index 0..15
    const int p0   = (tile & 255) << 4;                      // pixel base within image
    const float* xg = x + (size_t)img * CHW + p0 + ln;

    // ---- GEMM1: accumulate M, store into padded per-pixel LDS blocks ------
    for (int t = 0; t < 16; ++t) {                           // rk column tile (== r)
        v8f acc = {};
        const v2f* wb2 = (const v2f*)wp2 + t * 2048 + ln;    // pair-packed w_core
#pragma unroll 4
        for (int kb = 0; kb < 256; kb += 4) {
            const int c0 = kb + 2 * half;                    // even
            v2f a;
            a.x = xg[(size_t)c0 * HW];
            a.y = xg[(size_t)(c0 + 1) * HW];
            v2f b = wb2[(c0 >> 1) * 16];                     // {B[c0][n], B[c0+1][n]}
            acc = __builtin_amdgcn_wmma_f32_16x16x4_f32(false, a, false, b,
                                                        (short)0, acc, false, false);
        }
#pragma unroll
        for (int v = 0; v < 8; ++v) {
            const int px = v + 8 * half;                     // D row = pixel
            Mw[px * PIXS + t * MROW + ln] = acc[v];          // M_px[r=t][k=ln]
        }
    }

    // ---- per-pixel squaring: P = M @ M (4 WMMAs of K=4 each) --------------
    for (int px = 0; px < 16; ++px) {
        float* mp = Mw + px * PIXS;
        v8f acc = {};
#pragma unroll
        for (int kb = 0; kb < 16; kb += 4) {
            const int k0 = kb + 2 * half;
            v2f a, b;
            a.x = mp[ln * MROW + k0];                        // M[e=ln][f=k0]
            a.y = mp[ln * MROW + k0 + 1];
            b.x = mp[k0 * MROW + ln];                        // M[f=k0][g=ln]
            b.y = mp[(k0 + 1) * MROW + ln];
            acc = __builtin_amdgcn_wmma_f32_16x16x4_f32(false, a, false, b,
                                                        (short)0, acc, false, false);
        }
#pragma unroll
        for (int v = 0; v < 8; ++v)
            mp[(v + 8 * half) * MROW + ln] = acc[v];         // P[e][g] over M in place
    }

    // ---- GEMM2: Y = P * Vpack; transpose via Yw; 64B-contiguous stores ----
    const int n1 = lane >> 2;                                // 0..7  (d sub-row)
    const int pp = (lane & 3) << 2;                          // 0,4,8,12 (pixel sub-col)
    for (int td = 0; td < 16; ++td) {                        // output-channel tile
        v8f acc = {};
        const v2f* vb2 = (const v2f*)vp2 + td * 16 + ln;     // pair-packed V
#pragma unroll 4
        for (int kb = 0; kb < 256; kb += 4) {
            const int k0 = kb + 2 * half;                    // even; k index = e*16+g
            v2f a;
            const int blk = k0 >> 4, off = k0 & 15;          // padded P addressing
            a.x = Mw[ln * PIXS + blk * MROW + off];          // P[pixel=ln][k0]
            a.y = Mw[ln * PIXS + blk * MROW + off + 1];
            v2f b = vb2[(k0 >> 1) * 256];                    // {V[k0][d], V[k0+1][d]}
            acc = __builtin_amdgcn_wmma_f32_16x16x4_f32(false, a, false, b,
                                                        (short)0, acc, false, false);
        }
#pragma unroll
        for (int v = 0; v < 8; ++v)
            Yw[ln * YROW + v + 8 * half] = acc[v];           // Yw[d-col ln][pixel]

        const size_t ob = (size_t)img * CHW + (size_t)(td * 16) * HW + p0;
        float4 r1 = *(const float4*)&Yw[n1 * YROW + pp];
        float4 r2 = *(const float4*)&Yw[(n1 + 8) * YROW + pp];
        *(float4*)&out[ob + (size_t)n1 * HW + pp]       = r1;
        *(float4*)&out[ob + (size_t)(n1 + 8) * HW + pp] = r2;
    }
}

// ---------------------------------------------------------------------------
// InstanceNorm2d (biased var, eps=1e-5, no affine) + Softsign, in-place.
// One 256-thread workgroup per (a, d) plane of 4096 elements.
// ---------------------------------------------------------------------------
__global__ __launch_bounds__(256) void inorm_softsign(float* __restrict__ y) {
    __shared__ float bs[256];
    __shared__ float bss[256];
    __shared__ float mean_s, rs_s;

    float* p = y + (size_t)blockIdx.x * 4096;
    const int tid = threadIdx.x;

    float v[16];
    float s = 0.f, ss = 0.f;
#pragma unroll
    for (int i = 0; i < 16; ++i) {
        float t = p[tid + 256 * i];
        v[i] = t;
        s  += t;
        ss += t * t;
    }
    bs[tid] = s;
    bss[tid] = ss;
    __syncthreads();
    for (int off = 128; off > 0; off >>= 1) {
        if (tid < off) {
            bs[tid]  += bs[tid + off];
            bss[tid] += bss[tid + off];
        }
        __syncthreads();
    }
    if (tid == 0) {
        const float mean = bs[0] * (1.0f / 4096.0f);
        const float var  = bss[0] * (1.0f / 4096.0f) - mean * mean;
        mean_s = mean;
        rs_s   = rsqrtf(var + 1e-5f);
    }
    __syncthreads();
    const float mean = mean_s, rs = rs_s;
#pragma unroll
    for (int i = 0; i < 16; ++i) {
        const float yn = (v[i] - mean) * rs;
        p[tid + 256 * i] = yn / (1.0f + fabsf(yn));
    }
}

// ---------------------------------------------------------------------------
extern "C" void kernel_launch(void* const* d_in, const int* in_sizes, int n_in,
                              void* d_out, int out_size, void* d_ws, size_t ws_size,
                              hipStream_t stream) {
    (void)in_sizes; (void)n_in; (void)out_size; (void)ws_size;
    const float* x     = (const float*)d_in[0];
    const float* wcore = (const float*)d_in[1];
    const float* wout  = (const float*)d_in[2];
    float* out = (float*)d_out;
    float* ws  = (float*)d_ws;                   // 512 KB scratch: Wp2 | Vp2
    const float* wp2 = ws;
    const float* vp2 = ws + 65536;

    // 1) pair-pack both weight operands
    repack_weights<<<512, 256, 0, stream>>>(wcore, wout, ws);

    // 2) main tensor-ring pass (4096 tiles of 16 pixels, 2 waves per block)
    const size_t smem_bytes = (WAVES * (16 * PIXS) + WAVES * (16 * YROW)) * sizeof(float);
    tr_main<<<4096 / WAVES, WAVES * 32, smem_bytes, stream>>>(x, wp2, vp2, out);

    // 3) instance norm + softsign, one block per (a, d) plane
    inorm_softsign<<<16 * 256, 256, 0, stream>>>(out);
}